// BlockSparseLocallyConnected_30356828848771
// MI455X (gfx1250) — compile-verified
//
#include <hip/hip_runtime.h>

// ---------------------------------------------------------------------------
// BlockSparseLocallyConnected: B=128, IN_C=OUT_C=32, H=32, N=1024, BS=32
// out[b, r*32+co] = sum_{j: brow[j]==r} sum_ci x[b, bcol[j]*32+ci] * (W*M)[j,co,ci] + bias
// fp32 end-to-end via V_WMMA_F32_16X16X4_F32 (memory-bound; keeps precision).
// Double-buffered LDS pipeline; X tile staged with async global->LDS (b128).
// ---------------------------------------------------------------------------

typedef __attribute__((ext_vector_type(2))) float v2f;
typedef __attribute__((ext_vector_type(8))) float v8f;
typedef int v4i_vs __attribute__((vector_size(16)));   // matches builtin param
typedef v4i_vs __attribute__((address_space(1))) g_v4i;
typedef v4i_vs __attribute__((address_space(3))) l_v4i;

#define BATCH 128
#define NCH   32          // IN_C == OUT_C == BS
#define NSPAT 1024        // H*H
#define BS2   1024        // 32*32 block elements
#define LDSP  36          // padded row stride: 144B = 16B-aligned, conflict-free

#if __has_builtin(__builtin_amdgcn_global_load_async_to_lds_b128)
#define USE_ASYNC_LDS 1
#else
#define USE_ASYNC_LDS 0
#endif

// ---------------------------------------------------------------------------
// Kernel 1: input [B, C, N] -> ws_x [B, N, C]   (32x32 LDS tile transpose)
// ---------------------------------------------------------------------------
__global__ void bslc_transpose_in(const float* __restrict__ in,
                                  float* __restrict__ xT) {
  __shared__ float t[32][33];
  const int tid  = threadIdx.x;
  const int lane = tid & 31;
  const int row8 = tid >> 5;
  const int b    = blockIdx.y;
  const int s0   = blockIdx.x * 32;

#pragma unroll
  for (int i = 0; i < 4; ++i) {
    int cch = i * 8 + row8;
    t[cch][lane] = in[((size_t)b * NCH + cch) * NSPAT + s0 + lane];
  }
  __syncthreads();
#pragma unroll
  for (int i = 0; i < 4; ++i) {
    int sl = i * 8 + row8;
    xT[((size_t)b * NSPAT + (s0 + sl)) * NCH + lane] = t[lane][sl];
  }
}

// ---------------------------------------------------------------------------
// Kernel 3: ws_y [B, N, C] -> out [B, C, N]
// ---------------------------------------------------------------------------
__global__ void bslc_transpose_out(const float* __restrict__ yT,
                                   float* __restrict__ out) {
  __shared__ float t[32][33];
  const int tid  = threadIdx.x;
  const int lane = tid & 31;
  const int row8 = tid >> 5;
  const int b    = blockIdx.y;
  const int s0   = blockIdx.x * 32;

#pragma unroll
  for (int i = 0; i < 4; ++i) {
    int sl = i * 8 + row8;
    t[sl][lane] = yT[((size_t)b * NSPAT + (s0 + sl)) * NCH + lane];
  }
  __syncthreads();
#pragma unroll
  for (int i = 0; i < 4; ++i) {
    int cch = i * 8 + row8;
    out[((size_t)b * NCH + cch) * NSPAT + s0 + lane] = t[lane][cch];
  }
}

// ---------------------------------------------------------------------------
// Kernel 2: block-sparse GEMM via WMMA f32 16x16x4, double-buffered.
// grid: (1024 block rows, 4 batch tiles of 32), block: 128 (4 waves, wave32).
// ---------------------------------------------------------------------------
__device__ __forceinline__ int lb_search(const int* __restrict__ a, int n, int key) {
  int lo = 0, hi = n;
  while (lo < hi) {
    int mid = (lo + hi) >> 1;
    if (a[mid] < key) lo = mid + 1; else hi = mid;
  }
  return lo;
}

__global__ void bslc_bsr_wmma(const float* __restrict__ xT,      // [B, N, 32]
                              const float* __restrict__ weight,  // [nnz, 32, 32]
                              const float* __restrict__ mask,    // [nnz, 32, 32]
                              const float* __restrict__ bias,    // [32768]
                              const int*  __restrict__ brow,     // [nnz] sorted
                              const int*  __restrict__ bcol,     // [nnz]
                              int nnz,
                              float* __restrict__ yT) {          // [B, N, 32]
  __shared__ __align__(16) float Xs[2][32][LDSP];  // [buf][batch_local][cin]
  __shared__ __align__(16) float Vs[2][32][LDSP];  // [buf][cout][cin]

  const int tid   = threadIdx.x;
  const int wv    = tid >> 5;
  const int lane  = tid & 31;
  const int half  = lane >> 4;         // 0: lanes 0-15, 1: lanes 16-31
  const int l15   = lane & 15;
  const int mBase = (wv >> 1) * 16;    // batch sub-tile
  const int nBase = (wv & 1) * 16;     // cout sub-tile

  const int r      = blockIdx.x;       // output block row == spatial index
  const int btBase = blockIdx.y * 32;  // batch base

  const int start = lb_search(brow, nnz, r);
  const int end   = lb_search(brow, nnz, r + 1);

  float wr[8];
#if !USE_ASYNC_LDS
  float xr[8];
#endif

  v8f acc0 = {}, acc1 = {};

  // Start fetching block j into LDS buffer `buf` (X async, W*M into regs).
  auto stage_issue = [&](int j, int buf) {
    const int c = bcol[j];
#if USE_ASYNC_LDS
#pragma unroll
    for (int i = 0; i < 2; ++i) {
      int chunk = tid * 2 + i;                 // 256 x 16B chunks of the tile
      int row = chunk >> 3, c4 = (chunk & 7) * 4;
      g_v4i* g = (g_v4i*)&xT[((size_t)(btBase + row) * NSPAT + c) * NCH + c4];
      l_v4i* l = (l_v4i*)&Xs[buf][row][c4];
      __builtin_amdgcn_global_load_async_to_lds_b128(g, l, 0, 0);
    }
#else
#pragma unroll
    for (int i = 0; i < 8; ++i) {
      int idx = tid + i * 128;
      int row = idx >> 5, col = idx & 31;
      xr[i] = xT[((size_t)(btBase + row) * NSPAT + c) * NCH + col];
    }
#endif
#pragma unroll
    for (int i = 0; i < 8; ++i) {
      int idx = tid + i * 128;
      wr[i] = weight[(size_t)j * BS2 + idx] * mask[(size_t)j * BS2 + idx];
    }
  };

  // Finish staging into `buf` and make it visible to all waves.
  auto stage_commit = [&](int buf) {
#pragma unroll
    for (int i = 0; i < 8; ++i) {
      int idx = tid + i * 128;
      int row = idx >> 5, col = idx & 31;
      Vs[buf][row][col] = wr[i];
#if !USE_ASYNC_LDS
      Xs[buf][row][col] = xr[i];
#endif
    }
#if USE_ASYNC_LDS
#if __has_builtin(__builtin_amdgcn_s_wait_asynccnt)
    __builtin_amdgcn_s_wait_asynccnt(0);
#else
    asm volatile("s_wait_asynccnt 0x0" ::: "memory");
#endif
#endif
    __syncthreads();
  };

  if (start < end) {
    stage_issue(start, 0);
    stage_commit(0);
  }

  for (int j = start; j < end; ++j) {
    const int buf = (j - start) & 1;
    if (j + 1 < end) stage_issue(j + 1, buf ^ 1);
    if (j + 2 < end)
      __builtin_prefetch(&weight[(size_t)(j + 2) * BS2 + tid * 8], 0, 3);

    // Two independent accumulator chains (even/odd K-chunks) for WMMA ILP.
#pragma unroll
    for (int kk = 0; kk < 8; kk += 2) {
      const int k0 = kk * 4 + 2 * half;
      const int k1 = (kk + 1) * 4 + 2 * half;
      // A (16x4 f32): lanes 0-15 -> M=lane, K={0,1}; lanes 16-31 -> K={2,3}
      v2f a0 = { Xs[buf][mBase + l15][k0], Xs[buf][mBase + l15][k0 + 1] };
      v2f b0 = { Vs[buf][nBase + l15][k0], Vs[buf][nBase + l15][k0 + 1] };
      v2f a1 = { Xs[buf][mBase + l15][k1], Xs[buf][mBase + l15][k1 + 1] };
      v2f b1 = { Vs[buf][nBase + l15][k1], Vs[buf][nBase + l15][k1 + 1] };
      acc0 = __builtin_amdgcn_wmma_f32_16x16x4_f32(false, a0, false, b0,
                                                   (short)0, acc0, false, false);
      acc1 = __builtin_amdgcn_wmma_f32_16x16x4_f32(false, a1, false, b1,
                                                   (short)0, acc1, false, false);
    }

    if (j + 1 < end) stage_commit(buf ^ 1);
  }

  // D layout: VGPR v -> M = v + 8*half (batch), N = l15 (cout)
  v8f acc = acc0 + acc1;
  const float bv = bias[r * 32 + nBase + l15];
#pragma unroll
  for (int v = 0; v < 8; ++v) {
    const int b = btBase + mBase + 8 * half + v;
    yT[((size_t)b * NSPAT + r) * NCH + nBase + l15] = acc[v] + bv;
  }
}

// ---------------------------------------------------------------------------
extern "C" void kernel_launch(void* const* d_in, const int* in_sizes, int n_in,
                              void* d_out, int out_size, void* d_ws, size_t ws_size,
                              hipStream_t stream) {
  const float* input  = (const float*)d_in[0];
  const float* weight = (const float*)d_in[1];
  const float* mask   = (const float*)d_in[2];
  const float* bias   = (const float*)d_in[3];
  const int*   brow   = (const int*)d_in[4];
  const int*   bcol   = (const int*)d_in[5];
  float*       out    = (float*)d_out;

  const int nnz = in_sizes[1] / BS2;

  float* ws_x = (float*)d_ws;                              // [B, N, 32] = 16 MB
  float* ws_y = ws_x + (size_t)BATCH * NSPAT * NCH;        // [B, N, 32] = 16 MB

  dim3 gT(NSPAT / 32, BATCH);
  bslc_transpose_in<<<gT, 256, 0, stream>>>(input, ws_x);

  dim3 gM(NSPAT, BATCH / 32);
  bslc_bsr_wmma<<<gM, 128, 0, stream>>>(ws_x, weight, mask, bias, brow, bcol,
                                        nnz, ws_y);

  bslc_transpose_out<<<gT, 256, 0, stream>>>(ws_y, out);
}